// PointNetModule_30236569764418
// MI455X (gfx1250) — compile-verified
//
#include <hip/hip_runtime.h>

#define Bn   4
#define Nn   4096
#define Mn   2048
#define Kn   64
#define R2c  0.09f
#define H3n  128

typedef float v2f __attribute__((ext_vector_type(2)));
typedef float v8f __attribute__((ext_vector_type(8)));

__device__ __forceinline__ v8f v8f_zero() {
    v8f z = {0.f,0.f,0.f,0.f,0.f,0.f,0.f,0.f};
    return z;
}

// ---------------------------------------------------------------------------
// Kernel 1: farthest point sampling, one workgroup per batch cloud.
// All N=4096 positions + running min-distances live in registers
// (512 threads x 8 points). Each of the 2047 steps: register argmax ->
// wave32 shfl reduce -> 16-entry LDS reduce -> broadcast winner.
// Writes idx[] to workspace and cent[] into d_out's second output region.
// ---------------------------------------------------------------------------
__global__ __launch_bounds__(512)
void fps_kernel(const float* __restrict__ pos, float* __restrict__ cent,
                int* __restrict__ idx)
{
    const int b   = blockIdx.x;
    const int tid = threadIdx.x;
    const float* p = pos + (size_t)b * Nn * 3;

    float px[8], py[8], pz[8], dd[8];
    const float x0 = p[0], y0 = p[1], z0 = p[2];
#pragma unroll
    for (int k = 0; k < 8; ++k) {
        int i = tid + k * 512;
        float a = p[3*i], c1 = p[3*i+1], c2 = p[3*i+2];
        px[k] = a; py[k] = c1; pz[k] = c2;
        float dx = a - x0, dy = c1 - y0, dz = c2 - z0;
        dd[k] = dx*dx + dy*dy + dz*dz;
    }
    if (tid == 0) {
        idx[b * Mn] = 0;
        size_t co = (size_t)b * Mn * 3;
        cent[co] = x0; cent[co+1] = y0; cent[co+2] = z0;
    }

    __shared__ float redV[16];
    __shared__ int   redI[16];
    __shared__ float sCx, sCy, sCz;

    const int lane = tid & 31, wid = tid >> 5;

    for (int s = 1; s < Mn; ++s) {
        // thread-local argmax (strict > keeps lowest index: k ascending)
        float bv = -1.0f; int bi = 0;
#pragma unroll
        for (int k = 0; k < 8; ++k) {
            if (dd[k] > bv) { bv = dd[k]; bi = tid + k * 512; }
        }
        // wave32 reduce, tie-break lowest index (matches jnp.argmax)
#pragma unroll
        for (int off = 16; off > 0; off >>= 1) {
            float ov = __shfl_down(bv, off, 32);
            int   oi = __shfl_down(bi, off, 32);
            if (ov > bv || (ov == bv && oi < bi)) { bv = ov; bi = oi; }
        }
        if (lane == 0) { redV[wid] = bv; redI[wid] = bi; }
        __syncthreads();
        if (tid == 0) {
            float best = redV[0]; int bidx = redI[0];
            for (int t = 1; t < 16; ++t) {
                float v = redV[t]; int i2 = redI[t];
                if (v > best || (v == best && i2 < bidx)) { best = v; bidx = i2; }
            }
            float cx = p[3*bidx], cy = p[3*bidx+1], cz = p[3*bidx+2];
            sCx = cx; sCy = cy; sCz = cz;
            idx[b * Mn + s] = bidx;
            size_t co = ((size_t)b * Mn + s) * 3;
            cent[co] = cx; cent[co+1] = cy; cent[co+2] = cz;
        }
        __syncthreads();
        const float cx = sCx, cy = sCy, cz = sCz;
#pragma unroll
        for (int k = 0; k < 8; ++k) {
            float dx = px[k]-cx, dy = py[k]-cy, dz = pz[k]-cz;
            dd[k] = fminf(dd[k], dx*dx + dy*dy + dz*dz);
        }
    }
}

// ---------------------------------------------------------------------------
// Kernel 2: radius-ball query. One wave32 per centroid; slot 0 = centroid
// itself, remaining in-radius points appended via LDS atomic counter.
// Unfilled slots stay -1 (invalid).
// ---------------------------------------------------------------------------
__global__ __launch_bounds__(256)
void ball_kernel(const float* __restrict__ pos, const float* __restrict__ cent,
                 const int* __restrict__ idx, int* __restrict__ nbr)
{
    __shared__ int cnt[8];
    const int w    = threadIdx.x >> 5;
    const int lane = threadIdx.x & 31;
    const int gw   = blockIdx.x * 8 + w;        // centroid id 0..8191
    const int b    = gw >> 11;
    const int m    = gw & (Mn - 1);
    const float* p = pos + (size_t)b * Nn * 3;
    const size_t cm = (size_t)b * Mn + m;

    const float cx = cent[cm*3], cy = cent[cm*3+1], cz = cent[cm*3+2];
    const int   ci = idx[cm];
    int* out = nbr + cm * Kn;

    for (int s = lane; s < Kn; s += 32) out[s] = -1;
    if (lane == 0) { out[0] = ci; cnt[w] = 1; }
    __syncthreads();

    for (int i = lane; i < Nn; i += 32) {
        float dx = p[3*i]-cx, dy = p[3*i+1]-cy, dz = p[3*i+2]-cz;
        float d2 = dx*dx + dy*dy + dz*dz;
        if (d2 <= R2c && i != ci) {
            int slot = atomicAdd(&cnt[w], 1);
            if (slot < Kn) out[slot] = i;
        }
    }
}

// ---------------------------------------------------------------------------
// Kernel 3: PointConv MLP (6->64->64->128) + masked max-aggregation.
// One wave32 per centroid; per-wave LDS tile [64][66] holds the current
// layer activation matrix (rows = neighbor slots). All matmuls use
// V_WMMA_F32_16X16X4_F32 (exact f32 semantics).
//
// Fragment layouts (per ISA 7.12.2):
//   A (16x4 f32):  lane -> M = lane&15,  half-wave -> K pair (hg*2)
//   B (4x16 f32):  lane -> N = lane&15,  half-wave -> K pair
//   C/D (16x16):   vgpr v -> M = v + 8*(lane>>4), lane&15 -> N
// All boundary/validity handling uses clamped unconditional loads + selects
// (no exec-masked load branches).
// ---------------------------------------------------------------------------
#define CONV_WAVES 2

__global__ __launch_bounds__(CONV_WAVES * 32)
void conv_kernel(const float* __restrict__ x,   const float* __restrict__ pos,
                 const float* __restrict__ W1,  const float* __restrict__ b1,
                 const float* __restrict__ W2,  const float* __restrict__ b2,
                 const float* __restrict__ W3,  const float* __restrict__ b3,
                 const float* __restrict__ cent, const int* __restrict__ nbr,
                 float* __restrict__ out)
{
    __shared__ float scratch[CONV_WAVES][64 * 66];   // stride 66: bank-conflict-free A reads
    const int w    = threadIdx.x >> 5;
    const int lane = threadIdx.x & 31;
    const int ln   = lane & 15;     // N (or M) within a 16x16 tile
    const int hg   = lane >> 4;     // half-wave -> K-pair / row-group select
    const int gw   = blockIdx.x * CONV_WAVES + w;
    const int b    = gw >> 11;
    const int m    = gw & (Mn - 1);
    const size_t cm = (size_t)b * Mn + m;
    float* S = &scratch[w][0];

    const float cx = cent[cm*3], cy = cent[cm*3+1], cz = cent[cm*3+2];
    const int* nb = nbr + cm * Kn;

    // ---- gather neighbor features into S rows (cols 0..5; 6,7 zero-pad) ----
    // Invalid slots: clamp index to 0 (always a valid address), load
    // unconditionally, then zero via selects -- avoids exec-masked branches.
    unsigned mask0, mask1;
    {
        int j0 = nb[lane];
        int j1 = nb[lane + 32];
        bool v0 = (j0 >= 0), v1 = (j1 >= 0);
        mask0 = (unsigned)__ballot(v0);
        mask1 = (unsigned)__ballot(v1);
        int j0c = v0 ? j0 : 0;
        int j1c = v1 ? j1 : 0;

        {
            const float* xp = x   + ((size_t)b * Nn + j0c) * 3;
            const float* pp = pos + ((size_t)b * Nn + j0c) * 3;
            float a0 = xp[0], a1 = xp[1], a2 = xp[2];
            float r0v = pp[0]-cx, r1v = pp[1]-cy, r2v = pp[2]-cz;
            float* r0 = S + lane * 66;
            r0[0] = v0 ? a0  : 0.f;  r0[1] = v0 ? a1  : 0.f;  r0[2] = v0 ? a2  : 0.f;
            r0[3] = v0 ? r0v : 0.f;  r0[4] = v0 ? r1v : 0.f;  r0[5] = v0 ? r2v : 0.f;
            r0[6] = 0.f;             r0[7] = 0.f;
        }
        {
            const float* xp = x   + ((size_t)b * Nn + j1c) * 3;
            const float* pp = pos + ((size_t)b * Nn + j1c) * 3;
            float a0 = xp[0], a1 = xp[1], a2 = xp[2];
            float r0v = pp[0]-cx, r1v = pp[1]-cy, r2v = pp[2]-cz;
            float* r1 = S + (lane + 32) * 66;
            r1[0] = v1 ? a0  : 0.f;  r1[1] = v1 ? a1  : 0.f;  r1[2] = v1 ? a2  : 0.f;
            r1[3] = v1 ? r0v : 0.f;  r1[4] = v1 ? r1v : 0.f;  r1[5] = v1 ? r2v : 0.f;
            r1[6] = 0.f;             r1[7] = 0.f;
        }
    }

    // ---------------- layer 1: [64,6] @ [6,64] (K padded to 8) ----------------
    v8f acc[4][4];
#pragma unroll
    for (int mt = 0; mt < 4; ++mt)
#pragma unroll
        for (int nt = 0; nt < 4; ++nt) acc[mt][nt] = v8f_zero();

#pragma unroll
    for (int kc = 0; kc < 2; ++kc) {
        const int kb   = kc*4 + hg*2;
        // clamped (always-valid) rows for unconditional loads; zero via select
        const int kbc0 = (kb     < 6) ? kb     : 0;
        const int kbc1 = (kb + 1 < 6) ? kb + 1 : 0;
        v2f Bf[4];
#pragma unroll
        for (int nt = 0; nt < 4; ++nt) {
            float wx = W1[kbc0*64 + nt*16 + ln];
            float wy = W1[kbc1*64 + nt*16 + ln];
            wx = (kb     < 6) ? wx : 0.f;
            wy = (kb + 1 < 6) ? wy : 0.f;
            v2f t = {wx, wy}; Bf[nt] = t;
        }
#pragma unroll
        for (int mt = 0; mt < 4; ++mt) {
            const float* ar = S + (mt*16 + ln) * 66 + kb;
            v2f Af = {ar[0], ar[1]};
#pragma unroll
            for (int nt = 0; nt < 4; ++nt)
                acc[mt][nt] = __builtin_amdgcn_wmma_f32_16x16x4_f32(
                    false, Af, false, Bf[nt], (short)0, acc[mt][nt], false, false);
        }
    }
    // bias + relu, write h1 back to S (all reads of F are done)
#pragma unroll
    for (int nt = 0; nt < 4; ++nt) {
        const float bias = b1[nt*16 + ln];
#pragma unroll
        for (int mt = 0; mt < 4; ++mt)
#pragma unroll
            for (int v = 0; v < 8; ++v) {
                int row = mt*16 + hg*8 + v;
                S[row*66 + nt*16 + ln] = fmaxf(acc[mt][nt][v] + bias, 0.f);
            }
    }

    // ---------------- layer 2: [64,64] @ [64,64] ----------------
    v8f acc2[4][4];
#pragma unroll
    for (int mt = 0; mt < 4; ++mt)
#pragma unroll
        for (int nt = 0; nt < 4; ++nt) acc2[mt][nt] = v8f_zero();

    for (int kc = 0; kc < 16; ++kc) {
        const int kb = kc*4 + hg*2;
        v2f Bf[4];
#pragma unroll
        for (int nt = 0; nt < 4; ++nt) {
            v2f t = { W2[kb*64 + nt*16 + ln], W2[(kb+1)*64 + nt*16 + ln] };
            Bf[nt] = t;
        }
#pragma unroll
        for (int mt = 0; mt < 4; ++mt) {
            const float* ar = S + (mt*16 + ln) * 66 + kb;
            v2f Af = {ar[0], ar[1]};
#pragma unroll
            for (int nt = 0; nt < 4; ++nt)
                acc2[mt][nt] = __builtin_amdgcn_wmma_f32_16x16x4_f32(
                    false, Af, false, Bf[nt], (short)0, acc2[mt][nt], false, false);
        }
    }
#pragma unroll
    for (int nt = 0; nt < 4; ++nt) {
        const float bias = b2[nt*16 + ln];
#pragma unroll
        for (int mt = 0; mt < 4; ++mt)
#pragma unroll
            for (int v = 0; v < 8; ++v) {
                int row = mt*16 + hg*8 + v;
                S[row*66 + nt*16 + ln] = fmaxf(acc2[mt][nt][v] + bias, 0.f);
            }
    }

    // ------- layer 3: [64,64] @ [64,128], masked max over rows -------
    const size_t ob = cm * H3n;
    const float NEG_INF = -__builtin_inff();
    for (int nt = 0; nt < 8; ++nt) {
        v8f a3[4];
#pragma unroll
        for (int mt = 0; mt < 4; ++mt) a3[mt] = v8f_zero();

        for (int kc = 0; kc < 16; ++kc) {
            const int kb = kc*4 + hg*2;
            v2f Bf = { W3[kb*128 + nt*16 + ln], W3[(kb+1)*128 + nt*16 + ln] };
#pragma unroll
            for (int mt = 0; mt < 4; ++mt) {
                const float* ar = S + (mt*16 + ln) * 66 + kb;
                v2f Af = {ar[0], ar[1]};
                a3[mt] = __builtin_amdgcn_wmma_f32_16x16x4_f32(
                    false, Af, false, Bf, (short)0, a3[mt], false, false);
            }
        }
        const float bias = b3[nt*16 + ln];
        float mx = NEG_INF;
#pragma unroll
        for (int mt = 0; mt < 4; ++mt)
#pragma unroll
            for (int v = 0; v < 8; ++v) {
                int row = mt*16 + hg*8 + v;
                unsigned ok = (row < 32) ? ((mask0 >> row) & 1u)
                                         : ((mask1 >> (row - 32)) & 1u);
                float val = a3[mt][v] + bias;
                mx = fmaxf(mx, ok ? val : NEG_INF);
            }
        // combine the two half-wave row groups (same column n)
        mx = fmaxf(mx, __shfl_xor(mx, 16, 32));
        if (lane < 16) out[ob + nt*16 + lane] = mx;
    }
}

// ---------------------------------------------------------------------------
extern "C" void kernel_launch(void* const* d_in, const int* in_sizes, int n_in,
                              void* d_out, int out_size, void* d_ws, size_t ws_size,
                              hipStream_t stream)
{
    (void)in_sizes; (void)n_in; (void)out_size; (void)ws_size;
    const float* x   = (const float*)d_in[0];
    const float* pos = (const float*)d_in[1];
    const float* W1  = (const float*)d_in[2];
    const float* b1  = (const float*)d_in[3];
    const float* W2  = (const float*)d_in[4];
    const float* b2  = (const float*)d_in[5];
    const float* W3  = (const float*)d_in[6];
    const float* b3  = (const float*)d_in[7];

    float* out  = (float*)d_out;                       // [B,M,128]
    float* cent = out + (size_t)Bn * Mn * H3n;         // [B,M,3]

    int* idxWs = (int*)d_ws;                           // B*M ints
    int* nbrWs = idxWs + Bn * Mn;                      // B*M*K ints

    fps_kernel <<<Bn, 512, 0, stream>>>(pos, cent, idxWs);
    ball_kernel<<<(Bn * Mn) / 8, 256, 0, stream>>>(pos, cent, idxWs, nbrWs);
    conv_kernel<<<(Bn * Mn) / CONV_WAVES, CONV_WAVES * 32, 0, stream>>>(
        x, pos, W1, b1, W2, b2, W3, b3, cent, nbrWs, out);
}